// KeypointsRotoLoss_30726196035812
// MI455X (gfx1250) — compile-verified
//
#include <hip/hip_runtime.h>

typedef __attribute__((ext_vector_type(2))) float v2f;
typedef __attribute__((ext_vector_type(8))) float v8f;

#define W_R_C 10.0f
#define W_T_C 0.1f
#define ORIGIN_EPS_C 1e-3f

#define NBLOCKS 4096
#define TPB 256

__device__ __forceinline__ void joint_accum(float px, float py, float pz,
                                            float gx, float gy, float gz,
                                            float& accR, float& accT) {
    float np2 = px * px + py * py + pz * pz;
    float ng2 = gx * gx + gy * gy + gz * gz;
    float np = sqrtf(np2);
    float ng = sqrtf(ng2);
    float mask = (np >= ORIGIN_EPS_C && ng >= ORIGIN_EPS_C) ? 1.0f : 0.0f;

    float dx = px - gx, dy = py - gy, dz = pz - gz;
    float dist = sqrtf(dx * dx + dy * dy + dz * dz);

    float rnp = (np > 0.0f) ? 1.0f / np : 1.0f;
    float rng = (ng > 0.0f) ? 1.0f / ng : 1.0f;
    float ax = px * rnp, ay = py * rnp, az = pz * rnp;
    float bx = gx * rng, by = gy * rng, bz = gz * rng;

    float c = ax * bx + ay * by + az * bz;
    float vx = ay * bz - az * by;
    float vy = az * bx - ax * bz;
    float vz = ax * by - ay * bx;
    float vsq = vx * vx + vy * vy + vz * vz;

    float denom = 1.0f + c;
    if (fabsf(denom) < 1e-8f) denom = 1e-8f;
    // tr(R) = 3 - 2*|v|^2/denom  =>  cos = 1 - |v|^2/denom
    float cosv = 1.0f - vsq / denom;
    cosv = fminf(1.0f, fmaxf(-1.0f, cosv));
    float theta = acosf(cosv);

    accR += mask * theta;
    accT += mask * dist;
}

__global__ __launch_bounds__(TPB) void roto_partial_kernel(
    const float* __restrict__ pred, const float* __restrict__ gt,
    float* __restrict__ partial, long long nChunks) {
    float accR = 0.0f, accT = 0.0f;

    long long g = (long long)blockIdx.x * TPB + threadIdx.x;
    long long stride = (long long)gridDim.x * TPB;
    const float4* __restrict__ p4 = (const float4*)pred;
    const float4* __restrict__ g4 = (const float4*)gt;

    // Each chunk = 4 joints = 3 float4 per input array (fully coalesced b128 loads)
    for (long long c = g; c < nChunks; c += stride) {
        float4 pa = p4[3 * c + 0];
        float4 pb = p4[3 * c + 1];
        float4 pc = p4[3 * c + 2];
        float4 ga = g4[3 * c + 0];
        float4 gb = g4[3 * c + 1];
        float4 gc = g4[3 * c + 2];

        joint_accum(pa.x, pa.y, pa.z, ga.x, ga.y, ga.z, accR, accT);
        joint_accum(pa.w, pb.x, pb.y, ga.w, gb.x, gb.y, accR, accT);
        joint_accum(pb.z, pb.w, pc.x, gb.z, gb.w, gc.x, accR, accT);
        joint_accum(pc.y, pc.z, pc.w, gc.y, gc.z, gc.w, accR, accT);
    }

    __shared__ float sR[TPB];
    __shared__ float sT[TPB];
    sR[threadIdx.x] = accR;
    sT[threadIdx.x] = accT;
    __syncthreads();

    // Wave 0 (lanes 0..31, full EXEC) reduces 256+256 partials via chained
    // V_WMMA_F32_16X16X4_F32 with B = ones:  D = A*1 + D.
    // A-matrix rows 0..7  <- theta partials, rows 8..15 <- dist partials.
    // ISA 16x4 f32 A layout: lane = (k>=2)*16 + m ; VGPR = k&1.
    if (threadIdx.x < 32) {
        int lane = threadIdx.x;
        int m = lane & 15;
        int khalf = lane >> 4;  // 0 -> K={0,1}, 1 -> K={2,3}
        const float* __restrict__ src = (m < 8) ? sR : sT;
        int row = m & 7;

        v8f d = {};
        v2f ones;
        ones.x = 1.0f;
        ones.y = 1.0f;

        for (int it = 0; it < 8; ++it) {
            int base = it * 32 + row * 4 + khalf * 2;
            v2f a;
            a.x = src[base + 0];
            a.y = src[base + 1];
            d = __builtin_amdgcn_wmma_f32_16x16x4_f32(
                /*neg_a=*/false, a, /*neg_b=*/false, ones,
                /*c_mod=*/(short)0, d, /*reuse_a=*/false, /*reuse_b=*/false);
        }

        // Lane 0 holds D[0..7][0] in its 8 accumulator VGPRs  -> total theta sum.
        // Lane 16 holds D[8..15][0]                            -> total dist sum.
        float t = d[0] + d[1] + d[2] + d[3] + d[4] + d[5] + d[6] + d[7];
        if (lane == 0) partial[2 * (long long)blockIdx.x + 0] = t;
        if (lane == 16) partial[2 * (long long)blockIdx.x + 1] = t;
    }
}

__global__ __launch_bounds__(256) void roto_final_kernel(
    const float* __restrict__ partial, int nBlocks, float* __restrict__ out,
    float invB) {
    __shared__ float sR[256];
    __shared__ float sT[256];
    float aR = 0.0f, aT = 0.0f;
    for (int i = threadIdx.x; i < nBlocks; i += 256) {
        aR += partial[2 * i + 0];
        aT += partial[2 * i + 1];
    }
    sR[threadIdx.x] = aR;
    sT[threadIdx.x] = aT;
    __syncthreads();
    for (int off = 128; off > 0; off >>= 1) {
        if ((int)threadIdx.x < off) {
            sR[threadIdx.x] += sR[threadIdx.x + off];
            sT[threadIdx.x] += sT[threadIdx.x + off];
        }
        __syncthreads();
    }
    if (threadIdx.x == 0) {
        out[0] = (W_R_C * sR[0] + W_T_C * sT[0]) * invB;
    }
}

extern "C" void kernel_launch(void* const* d_in, const int* in_sizes, int n_in,
                              void* d_out, int out_size, void* d_ws,
                              size_t ws_size, hipStream_t stream) {
    const float* pred = (const float*)d_in[0];
    const float* gt = (const float*)d_in[1];
    float* out = (float*)d_out;
    float* partial = (float*)d_ws;  // NBLOCKS * 2 floats = 32 KB

    long long totalFloats = (long long)in_sizes[0];  // B * J * 3
    long long nJoints = totalFloats / 3;             // B * J
    long long nChunks = nJoints / 4;                 // 4 joints per chunk
    long long Bsz = nJoints / 32;                    // J = 32 per reference
    float invB = 1.0f / (float)Bsz;

    roto_partial_kernel<<<NBLOCKS, TPB, 0, stream>>>(pred, gt, partial, nChunks);
    roto_final_kernel<<<1, 256, 0, stream>>>(partial, NBLOCKS, out, invB);
}